// SHConv_51092930953850
// MI455X (gfx1250) — compile-verified
//
#include <hip/hip_runtime.h>

typedef __attribute__((ext_vector_type(2))) float v2f;
typedef __attribute__((ext_vector_type(8))) float v8f;

#define DEG_MAX 80
#define NDEG    81           // L+1
#define NPOS    6561         // (L+1)^2
#define CIN     64
#define COUT    64
#define BATCH   32
#define NCPT    17
#define INTERVAL 5           // INTERVAL2 == 5 as well for L=80

// ---------------------------------------------------------------------------
// Kernel 1: build interpolated per-degree weights wl[l][i][o]  (81*64*64 f32)
// w1 segments cover l = 0..74 (15 segments of 5), w2 covers l = 75..80.
// ---------------------------------------------------------------------------
__global__ void shconv_interp(const float* __restrict__ weight,
                              float* __restrict__ wl) {
    int tid = blockIdx.x * blockDim.x + threadIdx.x;
    if (tid >= NDEG * CIN * COUT) return;
    int o = tid & 63;
    int i = (tid >> 6) & 63;
    int l = tid >> 12;

    int c0; float t;
    if (l < (NCPT - 2) * INTERVAL) {              // l in [0,75)
        c0 = l / INTERVAL;
        t  = (float)(l - c0 * INTERVAL) * (1.0f / (float)INTERVAL);
    } else {                                      // tail segment, anchors 15,16
        c0 = NCPT - 2;
        t  = (float)(l - (NCPT - 2) * INTERVAL) * (1.0f / (float)INTERVAL);
    }
    // weight layout (in, out, ncpt, 1): idx = (i*COUT + o)*NCPT + c
    const float* wb = weight + ((i * COUT + o) * NCPT);
    float w0 = wb[c0];
    float w1 = wb[c0 + 1];
    wl[tid] = (1.0f - t) * w0 + t * w1;           // wl layout [l][i][o]
}

// ---------------------------------------------------------------------------
// Kernel 2: per-(degree, batch) GEMM with V_WMMA_F32_16X16X4_F32.
// Out[b, o, n0:n0+2l+1] = W_l^T (64x64) @ X[b, :, n0:n0+2l+1]
// WMMA tile: M = out-channel (16), N = position (16), K = in-channel (4/step).
// Block: 128 threads = 4 waves, wave w owns out channels [16w, 16w+16).
// ---------------------------------------------------------------------------
__global__ void __launch_bounds__(128)
shconv_wmma(const float* __restrict__ x,
            const float* __restrict__ wl,
            float* __restrict__ out) {
    const int l    = blockIdx.x;            // degree 0..80
    const int b    = blockIdx.y;            // batch 0..31
    const int lane = threadIdx.x & 31;
    const int wave = threadIdx.x >> 5;      // o-tile 0..3
    const int obase = wave * 16;
    const int col   = lane & 15;            // A row (o_local) == B/C column (n_local)
    const int khi   = (lane >> 4) << 1;     // k_local offset: lanes 16-31 hold K+2

    const int n_start = l * l;
    const int n_end   = n_start + 2 * l + 1;

    // Preload A = W_l^T tile (16 x 64) for this wave's out channels: 32 VGPRs,
    // reused across every n-tile of the run.
    // ISA A layout (32-bit, 16x4): lane m=lane&15, k = vgpr + 2*(lane>=16).
    v2f A[16];
    const float* wbl = wl + l * (CIN * COUT) + obase + col;
#pragma unroll
    for (int ks = 0; ks < 16; ++ks) {
        const int k = 4 * ks + khi;
        A[ks].x = wbl[(k + 0) * COUT];
        A[ks].y = wbl[(k + 1) * COUT];
    }

    const float* xb = x   + (size_t)b * CIN  * NPOS;
    float*       ob = out + (size_t)b * COUT * NPOS;

    const int ntiles = (2 * l + 1 + 15) >> 4;
    for (int nt = 0; nt < ntiles; ++nt) {
        const int  n     = n_start + nt * 16 + col;
        const bool inrun = (n < n_end);
        const int  nc    = inrun ? n : (NPOS - 1);   // clamp: keep loads in-bounds

        v8f c = {0.f, 0.f, 0.f, 0.f, 0.f, 0.f, 0.f, 0.f};
#pragma unroll
        for (int ks = 0; ks < 16; ++ks) {
            const int k = 4 * ks + khi;
            // B layout (32-bit, 4x16): col = lane&15, k = vgpr + 2*(lane>=16).
            float b0 = xb[(size_t)(k + 0) * NPOS + nc];
            float b1 = xb[(size_t)(k + 1) * NPOS + nc];
            v2f bm;
            bm.x = inrun ? b0 : 0.0f;   // branchless select; EXEC stays all-ones
            bm.y = inrun ? b1 : 0.0f;
            c = __builtin_amdgcn_wmma_f32_16x16x4_f32(
                    false, A[ks], false, bm, (short)0, c, false, false);
        }

        if (inrun) {
            // C layout: vgpr r -> row (o_local) = r + 8*(lane>=16), col = lane&15
            const int orow = obase + ((lane >> 4) << 3);
            float* op = ob + (size_t)orow * NPOS + n;
#pragma unroll
            for (int r = 0; r < 8; ++r)
                op[(size_t)r * NPOS] = c[r];
        }
    }
}

// ---------------------------------------------------------------------------
// Host launcher. d_in[0] = x (32*64*6561 f32), d_in[1] = weight (64*64*17 f32).
// d_out = (32, 64, 6561) f32. d_ws needs 81*64*64*4 = 1,327,104 bytes.
// ---------------------------------------------------------------------------
extern "C" void kernel_launch(void* const* d_in, const int* in_sizes, int n_in,
                              void* d_out, int out_size, void* d_ws, size_t ws_size,
                              hipStream_t stream) {
    const float* x      = (const float*)d_in[0];
    const float* weight = (const float*)d_in[1];
    float*       out    = (float*)d_out;
    float*       wl     = (float*)d_ws;

    const int total = NDEG * CIN * COUT;
    shconv_interp<<<(total + 255) / 256, 256, 0, stream>>>(weight, wl);
    shconv_wmma<<<dim3(NDEG, BATCH), 128, 0, stream>>>(x, wl, out);
}